// SubPixelUpscaling_45526653337787
// MI455X (gfx1250) — compile-verified
//
#include <hip/hip_runtime.h>
#include <hip/hip_bf16.h>

// SubPixelUpscaling (depth-to-space, r=2, NHWC, channel layout (C, ry, rx))
//
// Pure permutation: 1.07 GB total HBM traffic, 0 FLOPs.
// MI455X roofline: 1.073e9 B / 23.3e12 B/s ~= 46 us. Memory-bound; the only
// optimization levers are full-width coalesced VMEM and non-temporal cache
// hints (CDNA5 TH=NT policy) so the 1 GB stream does not pollute L2.
//
// Layout insight: input float4 at (pixel p, group c) == the 2x2 output block
// at channel c. So each thread does ONE b128 load (wave-contiguous 512 B) and
// FOUR b32 stores, each of which is wave-contiguous 128 B (consecutive lanes
// = consecutive c = consecutive output addresses). No cross-lane exchange is
// needed, so LDS / async-tensor staging would only add latency, not bandwidth.

typedef __attribute__((ext_vector_type(4))) float f32x4;

#define R        2
#define C_OUT    64u
#define H_IN     256u
#define W_IN     256u
#define BATCH    8u

__global__ __launch_bounds__(256) void subpixel_d2s_45526653337787_kernel(
    const float* __restrict__ in, float* __restrict__ out)
{
    const uint32_t tid = blockIdx.x * 256u + threadIdx.x;

    const uint32_t c = tid & (C_OUT - 1u);        // output channel 0..63
    const uint32_t p = tid >> 6;                  // flat input pixel index
    const uint32_t j = p & (W_IN - 1u);
    const uint32_t i = (p >> 8) & (H_IN - 1u);
    const uint32_t b = p >> 16;

    // Input: pixel p holds 256 floats; channels [c*4, c*4+3] = one float4.
    // Consecutive tid -> consecutive float4 -> global_load_b128, 512 B/wave,
    // non-temporal (streamed once, never reused).
    const f32x4* __restrict__ in4 = reinterpret_cast<const f32x4*>(in);
    const f32x4 v = __builtin_nontemporal_load(in4 + ((size_t)p * C_OUT + c));

    // Output base: out[b, 2i, 2j, c]. Component k = y*2 + x goes to
    // out[b, 2i+y, 2j+x, c]. Each of the 4 stores is contiguous across the
    // wave (consecutive c), i.e. one full 128 B line per wave per store.
    const size_t row_stride = (size_t)(W_IN * R) * C_OUT;        // 512*64 floats
    const size_t obase =
        ((size_t)(b * (H_IN * R) + 2u * i) * (W_IN * R) + 2u * j) * C_OUT + c;

    float* __restrict__ o = out + obase;
    __builtin_nontemporal_store(v.x, o);                         // y=0, x=0
    __builtin_nontemporal_store(v.y, o + C_OUT);                 // y=0, x=1
    __builtin_nontemporal_store(v.z, o + row_stride);            // y=1, x=0
    __builtin_nontemporal_store(v.w, o + row_stride + C_OUT);    // y=1, x=1
}

extern "C" void kernel_launch(void* const* d_in, const int* in_sizes, int n_in,
                              void* d_out, int out_size, void* d_ws, size_t ws_size,
                              hipStream_t stream) {
    (void)n_in; (void)d_ws; (void)ws_size; (void)out_size;

    const float* in = (const float*)d_in[0];
    float* out = (float*)d_out;

    // One thread per float4 of input: in_sizes[0] = 8*256*256*256 = 134217728
    // -> 33,554,432 threads -> 131,072 blocks of 256.
    const uint32_t n_threads = (uint32_t)in_sizes[0] / 4u;
    const uint32_t n_blocks  = n_threads / 256u;

    subpixel_d2s_45526653337787_kernel<<<n_blocks, 256, 0, stream>>>(in, out);
}